// VectorQuantizerEMA_46059229283265
// MI455X (gfx1250) — compile-verified
//
#include <hip/hip_runtime.h>
#include <hip/hip_bf16.h>
#include <math.h>

// Problem constants
#define NN      65536     // N = 64*32*32 flattened rows
#define DD      64        // embedding dim
#define KK      1024      // num embeddings
#define HW      1024      // H*W per (b,c) plane
#define BSTRIDE 65536     // C*H*W, per-batch stride in inputs / quantized_out

typedef float v2f __attribute__((ext_vector_type(2)));
typedef float v4f __attribute__((ext_vector_type(4)));
typedef float v8f __attribute__((ext_vector_type(8)));

// ---------------------------------------------------------------------------
// Kernel 0: zero atomic accumulators (counts, dw, lossAcc) + precompute ||e_k||^2
// ---------------------------------------------------------------------------
__global__ __launch_bounds__(256) void vq_prep_kernel(const float* __restrict__ emb,
                                                      float* __restrict__ zeroRegion, // 66561 floats
                                                      float* __restrict__ eNorm) {
    int g = blockIdx.x * 256 + threadIdx.x;
    if (g < 66561) zeroRegion[g] = 0.0f;
    if (g < KK) {
        float s = 0.0f;
        #pragma unroll
        for (int d = 0; d < DD; ++d) { float e = emb[(size_t)g * DD + d]; s += e * e; }
        eNorm[g] = s;
    }
}

// ---------------------------------------------------------------------------
// Kernel 1: fused distance-GEMM (fp32 WMMA) + argmin over K.
// Block: 256 threads = 8 waves; each wave owns 16 x-rows (WMMA N dim).
// GEMM mapping: D[16 codebook][16 rows] = A(emb 16x4) x B(x^T 4x16), K-dim = D=64
// in 16 steps of 4. B (x rows) preloaded into 32 VGPRs per lane, reused across
// all codebook tiles. Per outer iteration we stage TWO 16-row codebook tiles
// and run TWO independent WMMA accumulation chains (acc0/acc1) sharing B, so
// the XDL pipe always has an independent WMMA to issue (no serial-chain stall).
//
// Layout assumptions (CDNA5 ISA 7.12.2, f32 16x4 A and mirrored 4x16 B):
//   A vgpr j, lane l: emb[kbase + l%16][4s + j + 2*(l/16)]
//   B vgpr j, lane l: x  [row   l%16 ][4s + j + 2*(l/16)]
//   C vgpr v, lane l: D[m = v + 8*(l/16)][n = l%16]
// ---------------------------------------------------------------------------
#define ROWS2 128   // rows per block
#define T2    256   // threads per block
#define LS    65    // LDS row stride (bank-conflict-free)

__global__ __launch_bounds__(T2) void vq_argmin_kernel(const float* __restrict__ inputs,
                                                       const float* __restrict__ emb,
                                                       const float* __restrict__ eNorm,
                                                       int* __restrict__ idxOut) {
    __shared__ float lds_x[ROWS2 * LS];   // 33280 B : x tile [row][d]
    __shared__ float lds_e[32 * LS];      //  8320 B : two codebook tiles [k][d]
    __shared__ float lds_n[KK];           //  4096 B : ||e_k||^2

    const int tid = threadIdx.x;
    const int n0  = blockIdx.x * ROWS2;          // 128 | 1024, stays in one (b) plane
    const int b   = n0 / HW;
    const int hw0 = n0 % HW;
    const float* inB = inputs + (size_t)b * BSTRIDE;

    // Stage x tile, transposed read: channel-major global (coalesced 512B runs)
    for (int e = tid; e < ROWS2 * DD; e += T2) {
        int c = e >> 7;            // e / 128
        int i = e & 127;
        lds_x[i * LS + c] = inB[c * HW + hw0 + i];
    }
    for (int e = tid; e < KK; e += T2) lds_n[e] = eNorm[e];
    __syncthreads();

    const int wave = tid >> 5;
    const int lane = tid & 31;
    const int ln   = lane & 15;
    const int h2   = (lane >> 4) * 2;            // K-offset for upper half-wave
    const int hi8  = (lane >> 4) * 8;            // M-offset of C vgprs
    const int rowL = wave * 16 + ln;             // local x row this lane's B slot covers

    // Preload B operand (this wave's 16x64 x tile) into 32 VGPRs
    float breg[32];
    #pragma unroll
    for (int s = 0; s < 16; ++s) {
        breg[2 * s]     = lds_x[rowL * LS + 4 * s + h2];
        breg[2 * s + 1] = lds_x[rowL * LS + 4 * s + 1 + h2];
    }

    float bestDist = 3.4e38f;
    int   bestIdx  = 0;

    for (int kt = 0; kt < KK / 32; ++kt) {       // 32 iterations x 32 codebook rows
        __syncthreads();
        for (int e = tid; e < 32 * DD; e += T2) {   // stage 32x64 codebook rows
            int kk = e >> 6, dd = e & 63;
            lds_e[kk * LS + dd] = emb[(size_t)(kt * 32 + kk) * DD + dd];
        }
        __syncthreads();

        v8f acc0 = {0.f, 0.f, 0.f, 0.f, 0.f, 0.f, 0.f, 0.f};
        v8f acc1 = {0.f, 0.f, 0.f, 0.f, 0.f, 0.f, 0.f, 0.f};
        #pragma unroll
        for (int s = 0; s < 16; ++s) {
            v2f a0, a1, bb;
            a0.x = lds_e[ln * LS + 4 * s + h2];
            a0.y = lds_e[ln * LS + 4 * s + 1 + h2];
            a1.x = lds_e[(16 + ln) * LS + 4 * s + h2];
            a1.y = lds_e[(16 + ln) * LS + 4 * s + 1 + h2];
            bb.x = breg[2 * s];
            bb.y = breg[2 * s + 1];
            acc0 = __builtin_amdgcn_wmma_f32_16x16x4_f32(
                false, a0, false, bb, (short)0, acc0, false, false);
            acc1 = __builtin_amdgcn_wmma_f32_16x16x4_f32(
                false, a1, false, bb, (short)0, acc1, false, false);
        }
        // dist_m = ||e_m||^2 - 2*dot (||x||^2 is argmin-invariant)
        #pragma unroll
        for (int v = 0; v < 8; ++v) {
            int   m = kt * 32 + v + hi8;
            float d = lds_n[m] - 2.0f * acc0[v];
            if (d < bestDist) { bestDist = d; bestIdx = m; }
        }
        #pragma unroll
        for (int v = 0; v < 8; ++v) {
            int   m = kt * 32 + 16 + v + hi8;
            float d = lds_n[m] - 2.0f * acc1[v];
            if (d < bestDist) { bestDist = d; bestIdx = m; }
        }
    }

    // Lane l and l^16 hold disjoint candidate sets for the same row; combine.
    float od = __shfl_xor(bestDist, 16, 32);
    int   oi = __shfl_xor(bestIdx, 16, 32);
    if (od < bestDist || (od == bestDist && oi < bestIdx)) { bestDist = od; bestIdx = oi; }
    if (h2 == 0) idxOut[n0 + rowL] = bestIdx;
}

// ---------------------------------------------------------------------------
// Kernel 2: gather quantized rows, write quantized_out (B,C,H,W), accumulate
// dw (scatter atomics), counts, and commitment-loss partial sum.
// Block: 256 threads handle 128 rows.
// ---------------------------------------------------------------------------
__global__ __launch_bounds__(256) void vq_output_kernel(const float* __restrict__ inputs,
                                                        const float* __restrict__ emb,
                                                        const int* __restrict__ idx,
                                                        float* __restrict__ qOut,
                                                        float* __restrict__ counts,
                                                        float* __restrict__ dw,
                                                        float* __restrict__ lossAcc) {
    __shared__ float lds_q[ROWS2 * LS];   // 33280 B
    __shared__ int   lds_k[ROWS2];
    __shared__ float red[256];

    const int tid = threadIdx.x;
    const int n0  = blockIdx.x * ROWS2;
    const int b   = n0 / HW;
    const int hw0 = n0 % HW;
    const float* inB  = inputs + (size_t)b * BSTRIDE;
    float*       outB = qOut   + (size_t)b * BSTRIDE;

    if (tid < ROWS2) lds_k[tid] = idx[n0 + tid];
    __syncthreads();

    // Gather codebook rows into LDS (row-major reads of L2-resident emb)
    for (int e = tid; e < ROWS2 * DD; e += 256) {
        int i = e >> 6, c = e & 63;
        lds_q[i * LS + c] = emb[(size_t)lds_k[i] * DD + c];
    }
    __syncthreads();

    float lsum = 0.0f;
    for (int e = tid; e < ROWS2 * DD; e += 256) {
        int c = e >> 7;             // channel-major: coalesced global r/w
        int i = e & 127;
        float x = inB[c * HW + hw0 + i];
        float q = lds_q[i * LS + c];
        float t = q - x;
        lsum += t * t;
        outB[c * HW + hw0 + i] = q;                       // x + sg(q-x) == q
        atomicAdd(&dw[(size_t)lds_k[i] * DD + c], x);     // enc^T @ flat
    }
    if (tid < ROWS2) atomicAdd(&counts[lds_k[tid]], 1.0f);

    red[tid] = lsum;
    __syncthreads();
    for (int s = 128; s > 0; s >>= 1) {
        if (tid < s) red[tid] += red[tid + s];
        __syncthreads();
    }
    if (tid == 0) atomicAdd(lossAcc, red[0]);
}

// ---------------------------------------------------------------------------
// Kernel 3: one-hot encodings, 256 MB streamed with non-temporal 16B stores.
// One block per row; thread t covers k = 4t..4t+3.
// ---------------------------------------------------------------------------
__global__ __launch_bounds__(256) void vq_encodings_kernel(const int* __restrict__ idx,
                                                           float* __restrict__ enc) {
    const int n = blockIdx.x;
    const int t = threadIdx.x;
    const int k = idx[n];                  // uniform -> scalar load
    const int base = t * 4;
    v4f v = {0.f, 0.f, 0.f, 0.f};
    int r = k - base;
    if (r >= 0 && r < 4) v[r] = 1.0f;
    __builtin_nontemporal_store(v, (v4f*)(enc + (size_t)n * KK + base));
}

// ---------------------------------------------------------------------------
// Kernel 4: EMA state update + loss/perplexity. Single block, 1024 threads.
// ---------------------------------------------------------------------------
__global__ __launch_bounds__(1024) void vq_finalize_kernel(const float* __restrict__ emaCS,
                                                           const float* __restrict__ emaW,
                                                           const float* __restrict__ counts,
                                                           const float* __restrict__ dw,
                                                           const float* __restrict__ lossAcc,
                                                           float* __restrict__ outLoss,
                                                           float* __restrict__ outPerp,
                                                           float* __restrict__ outNCS,
                                                           float* __restrict__ outNEW,
                                                           float* __restrict__ outNEmb) {
    __shared__ float red[1024];
    const int t = threadIdx.x;

    float cnt = counts[t];
    float ncs = 0.99f * emaCS[t] + 0.01f * cnt;

    red[t] = ncs;
    __syncthreads();
    for (int s = 512; s > 0; s >>= 1) { if (t < s) red[t] += red[t + s]; __syncthreads(); }
    float ntot = red[0];
    __syncthreads();

    float norm = (ncs + 1e-5f) / (ntot + (float)KK * 1e-5f) * ntot;
    outNCS[t] = norm;

    float p = cnt / (float)NN;
    red[t] = p * logf(p + 1e-10f);
    __syncthreads();
    for (int s = 512; s > 0; s >>= 1) { if (t < s) red[t] += red[t + s]; __syncthreads(); }
    if (t == 0) {
        *outPerp = expf(-red[0]);
        *outLoss = 0.25f * (*lossAcc) / ((float)NN * (float)DD);
    }

    float inv = 1.0f / fmaxf(norm, 1e-5f);
    #pragma unroll
    for (int d = 0; d < DD; ++d) {
        float nw = 0.99f * emaW[(size_t)t * DD + d] + 0.01f * dw[(size_t)t * DD + d];
        outNEW[(size_t)t * DD + d]  = nw;
        outNEmb[(size_t)t * DD + d] = nw * inv;
    }
}

// ---------------------------------------------------------------------------
extern "C" void kernel_launch(void* const* d_in, const int* in_sizes, int n_in,
                              void* d_out, int out_size, void* d_ws, size_t ws_size,
                              hipStream_t stream) {
    (void)in_sizes; (void)n_in; (void)out_size; (void)ws_size;

    const float* inputs = (const float*)d_in[0];   // (64,64,32,32)
    const float* emb    = (const float*)d_in[1];   // (1024,64)
    const float* emaCS  = (const float*)d_in[2];   // (1024,)
    const float* emaW   = (const float*)d_in[3];   // (1024,64)

    // Output tuple, concatenated flat fp32:
    float* out      = (float*)d_out;
    float* outLoss  = out;                               // 1
    float* outQ     = out + 1;                           // 4194304
    float* outPerp  = out + 1 + 4194304;                 // 1
    float* outEnc   = out + 2 + 4194304;                 // 67108864
    float* outNCS   = outEnc + 67108864;                 // 1024
    float* outNEW   = outNCS + 1024;                     // 65536
    float* outNEmb  = outNEW + 65536;                    // 65536

    // Workspace layout (floats); zeroed region = [0, 66561)
    float* ws      = (float*)d_ws;
    float* counts  = ws;               // 1024
    float* dw      = ws + 1024;        // 65536
    float* lossAcc = ws + 66560;       // 1
    float* eNorm   = ws + 66561;       // 1024
    int*   idxBuf  = (int*)(ws + 68608); // 65536 ints

    vq_prep_kernel<<<(66561 + 255) / 256, 256, 0, stream>>>(emb, ws, eNorm);
    vq_argmin_kernel<<<NN / ROWS2, T2, 0, stream>>>(inputs, emb, eNorm, idxBuf);
    vq_output_kernel<<<NN / ROWS2, 256, 0, stream>>>(inputs, emb, idxBuf, outQ, counts, dw, lossAcc);
    vq_encodings_kernel<<<NN, 256, 0, stream>>>(idxBuf, outEnc);
    vq_finalize_kernel<<<1, 1024, 0, stream>>>(emaCS, emaW, counts, dw, lossAcc,
                                               outLoss, outPerp, outNCS, outNEW, outNEmb);
}